// HierarchicalPoseEncoder_66391604462155
// MI455X (gfx1250) — compile-verified
//
#include <hip/hip_runtime.h>
#include <math.h>

// ---------------------------------------------------------------------------
// HierarchicalPoseEncoder on MI455X (gfx1250), wave32 + V_WMMA_F32_16X16X4_F32
//
// Two-kernel pipeline (same stream, graph-safe, deterministic):
//   1) pad_weights_kernel: zero-pads W0/W1/W2/biases into d_ws so that the
//      main kernel's B-operand / bias loads are straight unpredicated b64/b32
//      loads (no exec-mask branches around the WMMAs).
//   2) hier_pose_kernel: one wave owns a 16-sample tile end-to-end.
//      Block = 128 threads = 4 independent waves -> zero barriers.
//      All GEMMs (288->6, 19->19, 19->6) run on the f32 WMMA unit with bias
//      folded into C; ReLU applied on D registers.
// ---------------------------------------------------------------------------

typedef float v2f __attribute__((ext_vector_type(2)));
typedef float v4f __attribute__((ext_vector_type(4)));
typedef float v8f __attribute__((ext_vector_type(8)));

#define NJ 24   // joints
#define ND 6    // feature dim
#define NF 19   // per-joint MLP width (9+3+1+6)

// float offsets inside d_ws for the padded weights
#define OFF_W0  0                        // [16][288]
#define OFF_W1  (OFF_W0 + 16*288)        // [24][32][20]
#define OFF_W2  (OFF_W1 + NJ*32*20)      // [24][16][20]
#define OFF_B1  (OFF_W2 + NJ*16*20)      // [24][32]
#define OFF_B2  (OFF_B1 + NJ*32)         // [24][16]
#define OFF_B0  (OFF_B2 + NJ*16)         // [16]
#define WS_FLOATS (OFF_B0 + 16)

__constant__ int c_parents[NJ] = {-1,0,0,0,1,2,3,4,5,6,7,8,9,9,9,12,13,14,16,17,18,19,20,21};

static __device__ __forceinline__ v8f wmma_f32(v2f a, v2f b, v8f c) {
  // D(16x16,f32) = A(16x4,f32) * B(4x16,f32) + C
  return __builtin_amdgcn_wmma_f32_16x16x4_f32(
      /*neg_a=*/false, a, /*neg_b=*/false, b,
      /*c_mod=*/(short)0, c, /*reuse_a=*/false, /*reuse_b=*/false);
}

// =================== prep: zero-pad weights into workspace ==================
__global__ __launch_bounds__(256)
void pad_weights_kernel(const float* __restrict__ W0, const float* __restrict__ b0,
                        const float* __restrict__ W1, const float* __restrict__ b1,
                        const float* __restrict__ W2, const float* __restrict__ b2,
                        float* __restrict__ ws)
{
  const int t = threadIdx.x;            // single block of 256 threads
  float* W0p = ws + OFF_W0;
  float* W1p = ws + OFF_W1;
  float* W2p = ws + OFF_W2;
  float* b1p = ws + OFF_B1;
  float* b2p = ws + OFF_B2;
  float* b0p = ws + OFF_B0;

  for (int i = t; i < 16 * 288; i += 256) {             // W0p[n][k], n pad ->16
    const int n = i / 288, k = i - n * 288;
    W0p[i] = (n < ND) ? W0[n * 288 + k] : 0.0f;
  }
  for (int i = t; i < NJ * 32 * 20; i += 256) {         // W1p[j][n][k]
    const int j = i / 640, r = i - j * 640, n = r / 20, k = r - n * 20;
    W1p[i] = (n < NF && k < NF) ? W1[(j * NF + n) * NF + k] : 0.0f;
  }
  for (int i = t; i < NJ * 16 * 20; i += 256) {         // W2p[j][n][k]
    const int j = i / 320, r = i - j * 320, n = r / 20, k = r - n * 20;
    W2p[i] = (n < ND && k < NF) ? W2[(j * ND + n) * NF + k] : 0.0f;
  }
  for (int i = t; i < NJ * 32; i += 256) {              // b1p[j][n]
    const int j = i / 32, n = i - j * 32;
    b1p[i] = (n < NF) ? b1[j * NF + n] : 0.0f;
  }
  for (int i = t; i < NJ * 16; i += 256) {              // b2p[j][n]
    const int j = i / 16, n = i - j * 16;
    b2p[i] = (n < ND) ? b2[j * ND + n] : 0.0f;
  }
  if (t < 16) b0p[t] = (t < ND) ? b0[t] : 0.0f;
}

// ============================== main kernel ================================
__global__ __launch_bounds__(128)
void hier_pose_kernel(const float* __restrict__ rots, const float* __restrict__ Jtrs,
                      const float* __restrict__ ws,   float* __restrict__ out)
{
  // Per-wave LDS regions (no cross-wave sharing):
  //   sfeat[w][m][0..143] : out[j] features (j*6+n), [144..149] : gfeat
  //   sx[w][m][0..19]     : A-staging tile (x, then reused for h), K padded to 20
  __shared__ __align__(16) float sfeat[4][16][152];
  __shared__ __align__(16) float sx[4][16][20];

  const float* W0p = ws + OFF_W0;
  const float* W1p = ws + OFF_W1;
  const float* W2p = ws + OFF_W2;
  const float* b1p = ws + OFF_B1;
  const float* b2p = ws + OFF_B2;
  const float* b0p = ws + OFF_B0;

  const int  lane = threadIdx.x & 31;
  const int  wave = threadIdx.x >> 5;
  const int  half = lane >> 4;     // K-half select for A/B operands
  const int  ln   = lane & 15;     // M for A-layout, N for B/C/D-layout
  const long s0   = ((long)blockIdx.x * 4 + wave) * 16;   // base sample of tile

  float* feat = &sfeat[wave][0][0];   // stride 152 per sample
  float* xb   = &sx[wave][0][0];      // stride 20  per sample

  // ======================= gfeat = [rots|Jtrs] @ W0^T + b0 ==================
  // M=16 samples, K=288 (72 wmma steps: 54 from rots, 18 from Jtrs), N=16.
  const float* arow_r = rots + (s0 + ln) * 216;   // each lane streams its row
  const float* arow_j = Jtrs + (s0 + ln) * 72;
  __builtin_prefetch(arow_r, 0, 3);               // global_prefetch_b8
  __builtin_prefetch(arow_j, 0, 3);

  v8f acc;
  {
    const float ci = b0p[ln];
    #pragma unroll
    for (int g = 0; g < 8; ++g) acc[g] = ci;
  }
  #pragma unroll 3
  for (int kk = 0; kk < 54; ++kk) {               // rots segment of K
    const int k = kk * 4 + half * 2;
    const v2f a = *(const v2f*)(arow_r + k);
    const v2f b = *(const v2f*)(W0p + ln * 288 + k);
    acc = wmma_f32(a, b, acc);
  }
  #pragma unroll 3
  for (int kk = 54; kk < 72; ++kk) {              // Jtrs segment of K
    const int k = kk * 4 + half * 2;
    const v2f a = *(const v2f*)(arow_j + (k - 216));
    const v2f b = *(const v2f*)(W0p + ln * 288 + k);
    acc = wmma_f32(a, b, acc);
  }
  if (ln < ND) {
    #pragma unroll
    for (int g = 0; g < 8; ++g)
      feat[(g + half * 8) * 152 + 144 + ln] = acc[g];   // gfeat -> slot 24
  }

  // ============================ 24-joint chain ===============================
  for (int j = 0; j < NJ; ++j) {
    const int p     = c_parents[j];
    const int fslot = (p < 0) ? 24 : p;

    // ---- build x tile (16 x 19, K-padded to 20) in LDS; lane m handles row m
    if (lane < 16) {
      const int  m = lane;
      const long s = s0 + m;
      const float* rp = rots + (s * NJ + j) * 9;
      const float* tp = Jtrs + (s * NJ + j) * 3;
      __align__(16) float xr[20];
      #pragma unroll
      for (int i = 0; i < 9; ++i) xr[i] = rp[i];
      const float jx = tp[0], jy = tp[1], jz = tp[2];
      xr[9] = jx; xr[10] = jy; xr[11] = jz;
      float px = 0.0f, py = 0.0f, pz = 0.0f;
      if (p >= 0) {
        const float* pp = Jtrs + (s * NJ + p) * 3;
        px = pp[0]; py = pp[1]; pz = pp[2];
      }
      const float dx = jx - px, dy = jy - py, dz = jz - pz;
      xr[12] = sqrtf(dx * dx + dy * dy + dz * dz);       // bone length
      #pragma unroll
      for (int i = 0; i < ND; ++i) xr[13 + i] = feat[m * 152 + fslot * ND + i];
      xr[19] = 0.0f;                                     // K pad
      #pragma unroll
      for (int i = 0; i < 20; i += 4)
        *(v4f*)(xb + m * 20 + i) = *(const v4f*)(xr + i);
    }
    asm volatile("s_wait_dscnt 0" ::: "memory");         // x tile visible to wave

    // ---- h = relu(x @ W1[j]^T + b1[j]) : N split into [0..15] and [16..18]
    v8f c0, c1;
    {
      const float bias0 = b1p[j * 32 + ln];
      const float bias1 = b1p[j * 32 + 16 + ln];         // zero-padded past 18
      #pragma unroll
      for (int g = 0; g < 8; ++g) { c0[g] = bias0; c1[g] = bias1; }
    }
    const float* w1j = W1p + j * 32 * 20;                // [32 n][20 k], padded
    #pragma unroll
    for (int kk = 0; kk < 5; ++kk) {
      const int k = kk * 4 + half * 2;
      const v2f a   = *(const v2f*)(xb + ln * 20 + k);   // A reused for both N-tiles
      const v2f bt0 = *(const v2f*)(w1j + ln * 20 + k);
      const v2f bt1 = *(const v2f*)(w1j + (16 + ln) * 20 + k);
      c0 = wmma_f32(a, bt0, c0);
      c1 = wmma_f32(a, bt1, c1);
    }

    // ---- ReLU on D regs, write h back into the (already consumed) x tile
    #pragma unroll
    for (int g = 0; g < 8; ++g) {
      const int m = g + half * 8;
      xb[m * 20 + ln] = fmaxf(c0[g], 0.0f);
      if (ln < 3) xb[m * 20 + 16 + ln] = fmaxf(c1[g], 0.0f);
      if (ln == 3) xb[m * 20 + 19] = 0.0f;               // K pad
    }
    asm volatile("s_wait_dscnt 0" ::: "memory");         // h tile visible to wave

    // ---- out[j] = h @ W2[j]^T + b2[j] : N=6 in one tile
    v8f c2;
    {
      const float bias2 = b2p[j * 16 + ln];              // zero-padded past 5
      #pragma unroll
      for (int g = 0; g < 8; ++g) c2[g] = bias2;
    }
    const float* w2j = W2p + j * 16 * 20;                // [16 n][20 k], padded
    #pragma unroll
    for (int kk = 0; kk < 5; ++kk) {
      const int k = kk * 4 + half * 2;
      const v2f a  = *(const v2f*)(xb + ln * 20 + k);
      const v2f bt = *(const v2f*)(w2j + ln * 20 + k);
      c2 = wmma_f32(a, bt, c2);
    }
    if (ln < ND) {
      #pragma unroll
      for (int g = 0; g < 8; ++g)
        feat[(g + half * 8) * 152 + j * ND + ln] = c2[g];
    }
  }

  // ================= coalesced b128 streaming store of (16 x 144) ===========
  asm volatile("s_wait_dscnt 0" ::: "memory");
  #pragma unroll 4
  for (int idx = lane; idx < 16 * 36; idx += 32) {       // 36 float4 per sample row
    const int m = idx / 36;
    const int q = idx - m * 36;
    *(v4f*)(out + (s0 + m) * 144 + q * 4) = *(const v4f*)(feat + m * 152 + q * 4);
  }
}

extern "C" void kernel_launch(void* const* d_in, const int* in_sizes, int n_in,
                              void* d_out, int out_size, void* d_ws, size_t ws_size,
                              hipStream_t stream) {
  const float* rots = (const float*)d_in[0];
  const float* Jtrs = (const float*)d_in[1];
  const float* W0   = (const float*)d_in[2];
  const float* b0   = (const float*)d_in[3];
  const float* W1   = (const float*)d_in[4];
  const float* b1   = (const float*)d_in[5];
  const float* W2   = (const float*)d_in[6];
  const float* b2   = (const float*)d_in[7];
  float* out = (float*)d_out;
  float* ws  = (float*)d_ws;

  // 1) pad weights into workspace (~113 KB), 2) main kernel; same stream.
  hipLaunchKernelGGL(pad_weights_kernel, dim3(1), dim3(256), 0, stream,
                     W0, b0, W1, b1, W2, b2, ws);

  const long bsz    = (long)in_sizes[0] / (NJ * 9);   // rots is (B, 24, 9)
  const int  blocks = (int)(bsz / 64);                // 4 waves x 16 samples per block
  if (blocks > 0) {
    hipLaunchKernelGGL(hier_pose_kernel, dim3(blocks), dim3(128), 0, stream,
                       rots, Jtrs, ws, out);
  }
}